// NCRFAE_14061722927289
// MI455X (gfx1250) — compile-verified
//
#include <hip/hip_runtime.h>
#include <hip/hip_bf16.h>
#include <math.h>

// Problem constants (from reference)
#define B_   8
#define L_   64
#define WL_  32
#define E_   256
#define H_   512
#define H2_  256
#define G_   1024   // 4*H2
#define R_   128
#define N_   64     // = L
#define NEGC (-1000000000.0f)

typedef __attribute__((ext_vector_type(2))) float v2f;
typedef __attribute__((ext_vector_type(8))) float v8f;

__device__ __forceinline__ float sigf(float x) { return 1.0f / (1.0f + expf(-x)); }

// ---------------------------------------------------------------------------
// 1) embeds[b,l,:] = sum_w mask*emb[sents[b,l,w],:] / (elen + (elen==0))
//    grid: B*L blocks, E threads
// ---------------------------------------------------------------------------
__global__ void k_embed_mean(const int* __restrict__ sents, const float* __restrict__ mask,
                             const float* __restrict__ emb, float* __restrict__ embeds) {
  int bl = blockIdx.x;
  int d  = threadIdx.x;
  const int*   srow = sents + bl * WL_;
  const float* mrow = mask  + bl * WL_;
  float acc = 0.f, elen = 0.f;
  for (int w = 0; w < WL_; ++w) {
    float m = mrow[w];
    elen += m;
    acc  += m * emb[(long)srow[w] * E_ + d];
  }
  float denom = elen + (elen == 0.f ? 1.f : 0.f);
  embeds[(long)bl * E_ + d] = acc / denom;
}

// ---------------------------------------------------------------------------
// 2) Generic f32 WMMA GEMM:  C[m,n] = sum_k A[m,k]*W[n,k] + bias1[n] + bias2[n]
//    One wave per 16x16 C tile, K in steps of 4 (V_WMMA_F32_16X16X4_F32).
//    Optional batching via blockIdx.y strides.
// ---------------------------------------------------------------------------
__global__ void k_gemm_wmma(const float* __restrict__ A, int lda, long strideA,
                            const float* __restrict__ Wt, int ldw, long strideW,
                            float* __restrict__ C, int ldc, long strideC,
                            int M, int N, int K,
                            const float* __restrict__ bias1, const float* __restrict__ bias2) {
  int wave = threadIdx.x >> 5;
  int lane = threadIdx.x & 31;
  int half = lane >> 4;
  int l16  = lane & 15;
  int tilesN = N >> 4;
  int tilesM = M >> 4;
  int tile = blockIdx.x * (blockDim.x >> 5) + wave;
  if (tile >= tilesM * tilesN) return;
  int m0 = (tile / tilesN) << 4;
  int n0 = (tile % tilesN) << 4;
  const float* Ab = A  + (long)blockIdx.y * strideA;
  const float* Wb = Wt + (long)blockIdx.y * strideW;
  float*       Cb = C  + (long)blockIdx.y * strideC;

  const float* arow = Ab + (long)(m0 + l16) * lda + 2 * half;
  const float* wrow = Wb + (long)(n0 + l16) * ldw + 2 * half;
  v8f acc = {};
  for (int k = 0; k < K; k += 4) {
    v2f a, b;
    a.x = arow[k]; a.y = arow[k + 1];
    b.x = wrow[k]; b.y = wrow[k + 1];
    acc = __builtin_amdgcn_wmma_f32_16x16x4_f32(false, a, false, b, (short)0, acc, false, false);
  }
  int col = n0 + l16;
  float bias = 0.f;
  if (bias1) bias += bias1[col];
  if (bias2) bias += bias2[col];
#pragma unroll
  for (int r = 0; r < 8; ++r) {
    int row = m0 + r + 8 * half;
    Cb[(long)row * ldc + col] = acc[r] + bias;
  }
}

// ---------------------------------------------------------------------------
// 3) BiLSTM recurrence. NOTE: the reference scans over the *batch* axis (B=8
//    sequential steps), each step is (64 x 256) @ (256 x 1024) + cell update.
//    One 1024-thread block (32 waves) per direction. h state (64x256 f32,
//    64KB) lives in LDS as the WMMA A matrix; c state lives in registers.
//    Each wave owns 2 (row-tile, hidden-tile) pairs and computes the four
//    gate blocks (cols j0, 256+j0, 512+j0, 768+j0) whose C fragments align
//    element-wise, making the LSTM cell update pure per-lane register math.
// ---------------------------------------------------------------------------
__global__ __launch_bounds__(1024)
void k_bilstm(const float* __restrict__ Xf, const float* __restrict__ Xb,
              const float* __restrict__ Whh_f, const float* __restrict__ Whh_b,
              float* __restrict__ hout) {
  __shared__ float hS[64][H2_];   // 64 KB: h carry, rows = "L" index
  int dir = blockIdx.x;
  const float* Xp  = dir ? Xb    : Xf;
  const float* Whh = dir ? Whh_b : Whh_f;
  int tid  = threadIdx.x;
  int wave = tid >> 5, lane = tid & 31, half = lane >> 4, l16 = lane & 15;

  for (int idx = tid; idx < 64 * H2_; idx += 1024)
    hS[idx >> 8][idx & 255] = 0.f;

  // 64 tiles: 4 row-tiles x 16 hidden-unit tiles; 2 per wave
  int t0  = wave * 2;
  int m0a = ((t0)     >> 4) << 4, j0a = ((t0)     & 15) << 4;
  int m0b = ((t0 + 1) >> 4) << 4, j0b = ((t0 + 1) & 15) << 4;
  v8f cstate[2] = {};
  __syncthreads();

  for (int s = 0; s < B_; ++s) {
    int bstep = dir ? (B_ - 1 - s) : s;
    v8f hn[2];
#pragma unroll
    for (int tt = 0; tt < 2; ++tt) {
      int m0 = tt ? m0b : m0a;
      int j0 = tt ? j0b : j0a;
      v8f aI = {}, aF = {}, aG = {}, aO = {};
      const float* arow = &hS[m0 + l16][2 * half];
      const float* wI = Whh + (long)(j0 + l16)           * H2_ + 2 * half;
      const float* wF = Whh + (long)(H2_ + j0 + l16)     * H2_ + 2 * half;
      const float* wG = Whh + (long)(2 * H2_ + j0 + l16) * H2_ + 2 * half;
      const float* wO = Whh + (long)(3 * H2_ + j0 + l16) * H2_ + 2 * half;
      for (int k = 0; k < H2_; k += 4) {
        v2f a, b;
        a.x = arow[k]; a.y = arow[k + 1];
        b.x = wI[k]; b.y = wI[k + 1];
        aI = __builtin_amdgcn_wmma_f32_16x16x4_f32(false, a, false, b, (short)0, aI, false, false);
        b.x = wF[k]; b.y = wF[k + 1];
        aF = __builtin_amdgcn_wmma_f32_16x16x4_f32(false, a, false, b, (short)0, aF, false, false);
        b.x = wG[k]; b.y = wG[k + 1];
        aG = __builtin_amdgcn_wmma_f32_16x16x4_f32(false, a, false, b, (short)0, aG, false, false);
        b.x = wO[k]; b.y = wO[k + 1];
        aO = __builtin_amdgcn_wmma_f32_16x16x4_f32(false, a, false, b, (short)0, aO, false, false);
      }
      int j = j0 + l16;
#pragma unroll
      for (int r = 0; r < 8; ++r) {
        int row  = m0 + r + 8 * half;             // "L" position
        long xr  = ((long)bstep * L_ + row) * G_;
        float ig = sigf (aI[r] + Xp[xr + j]);
        float fg = sigf (aF[r] + Xp[xr + H2_ + j]);
        float gg = tanhf(aG[r] + Xp[xr + 2 * H2_ + j]);
        float og = sigf (aO[r] + Xp[xr + 3 * H2_ + j]);
        float c  = fg * cstate[tt][r] + ig * gg;
        cstate[tt][r] = c;
        hn[tt][r] = og * tanhf(c);
      }
    }
    __syncthreads();   // everyone done reading old hS
#pragma unroll
    for (int tt = 0; tt < 2; ++tt) {
      int m0 = tt ? m0b : m0a;
      int j  = (tt ? j0b : j0a) + l16;
#pragma unroll
      for (int r = 0; r < 8; ++r) {
        int row = m0 + r + 8 * half;
        float hv = hn[tt][r];
        hS[row][j] = hv;
        hout[((long)bstep * L_ + row) * H_ + dir * H2_ + j] = hv;
      }
    }
    __syncthreads();   // new hS visible before next step's WMMAs
  }
}

// ---------------------------------------------------------------------------
// 4) u[b,i] = h[i]·Ws[0:512] + h[i-1]·Ws[512:1024] + h[i+1]·Ws[1024:1536]
//    v[b,i] = same with Ws offset +1536.  grid: B*L blocks, 128 threads.
// ---------------------------------------------------------------------------
__global__ void k_uv(const float* __restrict__ h, const float* __restrict__ Ws,
                     float* __restrict__ u, float* __restrict__ v) {
  __shared__ float su[128], sv[128];
  int bl = blockIdx.x;
  int i  = bl & 63;
  int tid = threadIdx.x;
  const float* hc = h + (long)bl * H_;
  const float* hm = (i > 0)      ? h + (long)(bl - 1) * H_ : nullptr;
  const float* hp = (i < N_ - 1) ? h + (long)(bl + 1) * H_ : nullptr;
  float au = 0.f, av = 0.f;
  for (int d = tid; d < H_; d += 128) {
    float hv = hc[d];
    au += hv * Ws[d];
    av += hv * Ws[3 * H_ + d];
    if (hm) { float x = hm[d]; au += x * Ws[H_ + d];     av += x * Ws[4 * H_ + d]; }
    if (hp) { float x = hp[d]; au += x * Ws[2 * H_ + d]; av += x * Ws[5 * H_ + d]; }
  }
  su[tid] = au; sv[tid] = av;
  __syncthreads();
  for (int ofs = 64; ofs > 0; ofs >>= 1) {
    if (tid < ofs) { su[tid] += su[tid + ofs]; sv[tid] += sv[tid + ofs]; }
    __syncthreads();
  }
  if (tid == 0) { u[bl] = su[0]; v[bl] = sv[0]; }
}

// ---------------------------------------------------------------------------
// 5) crf[b,i,j] = u+v+bs ; jp = crf + log_softmax_j(S) + prior/n
//    grid: B*L blocks (one row), 64 threads.
// ---------------------------------------------------------------------------
__global__ void k_jp(const float* __restrict__ S, const float* __restrict__ u,
                     const float* __restrict__ v, const float* __restrict__ bs,
                     const float* __restrict__ prior,
                     float* __restrict__ crf, float* __restrict__ jp) {
  __shared__ float red[64];
  int bi = blockIdx.x;
  int b  = bi >> 6;
  int j  = threadIdx.x;
  float sc = S[(long)bi * N_ + j];
  red[j] = sc; __syncthreads();
  for (int ofs = 32; ofs > 0; ofs >>= 1) {
    if (j < ofs) red[j] = fmaxf(red[j], red[j + ofs]);
    __syncthreads();
  }
  float mx = red[0]; __syncthreads();
  red[j] = expf(sc - mx); __syncthreads();
  for (int ofs = 32; ofs > 0; ofs >>= 1) {
    if (j < ofs) red[j] += red[j + ofs];
    __syncthreads();
  }
  float lse = mx + logf(red[0]);
  float c = u[bi] + v[(b << 6) + j] + bs[0];
  crf[(long)bi * N_ + j] = c;
  jp [(long)bi * N_ + j] = c + (sc - lse) + prior[(long)bi * N_ + j] * (1.0f / (float)N_);
}

// ---------------------------------------------------------------------------
// 6) Eisner inside log-partition per batch. 4 DP tables in LDS (64KB).
//    grid: B blocks, 256 threads; serial loop over span width w.
// ---------------------------------------------------------------------------
__global__ void k_eisner(const float* __restrict__ crf, float* __restrict__ logZ) {
  __shared__ float Cr[N_ * N_], Cl[N_ * N_], Ir[N_ * N_], Il[N_ * N_];
  int b = blockIdx.x, tid = threadIdx.x;
  const float* s = crf + (long)b * N_ * N_;
  for (int idx = tid; idx < N_ * N_; idx += blockDim.x) {
    int i = idx >> 6, j = idx & 63;
    float dv = (i == j) ? 0.f : NEGC;
    Cr[idx] = dv; Cl[idx] = dv; Ir[idx] = NEGC; Il[idx] = NEGC;
  }
  __syncthreads();
  for (int w = 1; w < N_; ++w) {
    int ns = N_ - w;
    for (int sp = tid; sp < ns; sp += blockDim.x) {     // phase 1: I tables
      int i = sp, j = i + w;
      float m = -INFINITY, ss = 0.f;
      for (int k = i; k < j; ++k) {
        float t = Cr[i * N_ + k] + Cl[(k + 1) * N_ + j];
        if (t > m) { ss = ss * expf(m - t) + 1.f; m = t; } else ss += expf(t - m);
      }
      float inc = m + logf(ss);
      Ir[i * N_ + j] = inc + s[i * N_ + j];
      Il[i * N_ + j] = inc + s[j * N_ + i];
    }
    __syncthreads();
    for (int task = tid; task < 2 * ns; task += blockDim.x) {  // phase 2: C tables
      int which = task >= ns;
      int i = which ? task - ns : task;
      int j = i + w;
      float m = -INFINITY, ss = 0.f;
      if (!which) {
        for (int k = i + 1; k <= j; ++k) {
          float t = Ir[i * N_ + k] + Cr[k * N_ + j];
          if (t > m) { ss = ss * expf(m - t) + 1.f; m = t; } else ss += expf(t - m);
        }
        Cr[i * N_ + j] = m + logf(ss);
      } else {
        for (int k = i; k < j; ++k) {
          float t = Cl[i * N_ + k] + Il[k * N_ + j];
          if (t > m) { ss = ss * expf(m - t) + 1.f; m = t; } else ss += expf(t - m);
        }
        Cl[i * N_ + j] = m + logf(ss);
      }
    }
    __syncthreads();
  }
  if (tid == 0) logZ[b] = Cr[N_ - 1];   // Cr[0][n-1]
}

// ---------------------------------------------------------------------------
// 7) best[b] = logsumexp_t jp[b, heads[b,t], t] (t=1..n-1);
//    out = -mean(best - logZ)
// ---------------------------------------------------------------------------
__global__ void k_final(const float* __restrict__ jp, const int* __restrict__ heads,
                        const float* __restrict__ logZ, float* __restrict__ out) {
  __shared__ float part[B_];
  int b = threadIdx.x;
  if (b < B_) {
    const float* jb = jp + (long)b * N_ * N_;
    const int*   hb = heads + b * L_;
    float m = -INFINITY, ss = 0.f;
    for (int t = 1; t < N_; ++t) {
      float g = jb[(long)hb[t] * N_ + t];
      if (g > m) { ss = ss * expf(m - g) + 1.f; m = g; } else ss += expf(g - m);
    }
    part[b] = (m + logf(ss)) - logZ[b];
  }
  __syncthreads();
  if (threadIdx.x == 0) {
    float sum = 0.f;
    for (int x = 0; x < B_; ++x) sum += part[x];
    out[0] = -sum / (float)B_;
  }
}

// ---------------------------------------------------------------------------
extern "C" void kernel_launch(void* const* d_in, const int* in_sizes, int n_in,
                              void* d_out, int out_size, void* d_ws, size_t ws_size,
                              hipStream_t stream) {
  const int*   sents = (const int*)  d_in[0];
  const float* mask  = (const float*)d_in[1];
  const float* prior = (const float*)d_in[2];
  const int*   heads = (const int*)  d_in[3];
  const float* emb   = (const float*)d_in[4];
  const float* Wih_f = (const float*)d_in[5];
  const float* Whh_f = (const float*)d_in[6];
  const float* bih_f = (const float*)d_in[7];
  const float* bhh_f = (const float*)d_in[8];
  const float* Wih_b = (const float*)d_in[9];
  const float* Whh_b = (const float*)d_in[10];
  const float* bih_b = (const float*)d_in[11];
  const float* bhh_b = (const float*)d_in[12];
  const float* W1    = (const float*)d_in[13];
  const float* b1    = (const float*)d_in[14];
  const float* W2    = (const float*)d_in[15];
  const float* b2    = (const float*)d_in[16];
  const float* Ws    = (const float*)d_in[17];
  const float* bs    = (const float*)d_in[18];

  float* ws     = (float*)d_ws;
  float* embeds = ws;                  // 512*256      = 131072
  float* Xf     = embeds + 131072;     // 512*1024     = 524288
  float* Xb     = Xf + 524288;         // 524288
  float* hbuf   = Xb + 524288;         // 512*512      = 262144
  float* u      = hbuf + 262144;       // 512
  float* v      = u + 512;             // 512
  float* crf    = v + 512;             // 8*64*64      = 32768
  float* p1     = crf + 32768;         // 512*128      = 65536
  float* p2     = p1 + 65536;          // 65536
  float* S      = p2 + 65536;          // 32768
  float* jp     = S + 32768;           // 32768
  float* logZ   = jp + 32768;          // 8

  // 1) embedding mean
  k_embed_mean<<<B_ * L_, E_, 0, stream>>>(sents, mask, emb, embeds);

  // 2) input-gate GEMMs (M=512, N=1024, K=256), bias = bih + bhh
  {
    int blocks = ((512 / 16) * (1024 / 16)) / 4;   // 512
    k_gemm_wmma<<<dim3(blocks, 1), 128, 0, stream>>>(embeds, 256, 0, Wih_f, 256, 0,
                                                     Xf, 1024, 0, 512, 1024, 256, bih_f, bhh_f);
    k_gemm_wmma<<<dim3(blocks, 1), 128, 0, stream>>>(embeds, 256, 0, Wih_b, 256, 0,
                                                     Xb, 1024, 0, 512, 1024, 256, bih_b, bhh_b);
  }

  // 3) recurrence over the batch axis (8 steps), both directions in parallel
  k_bilstm<<<2, 1024, 0, stream>>>(Xf, Xb, Whh_f, Whh_b, hbuf);

  // 4) p1 = h@W1.T + b1, p2 = h@W2.T + b2  (M=512, N=128, K=512)
  {
    int blocks = ((512 / 16) * (128 / 16)) / 4;    // 64
    k_gemm_wmma<<<dim3(blocks, 1), 128, 0, stream>>>(hbuf, 512, 0, W1, 512, 0,
                                                     p1, 128, 0, 512, 128, 512, b1, nullptr);
    k_gemm_wmma<<<dim3(blocks, 1), 128, 0, stream>>>(hbuf, 512, 0, W2, 512, 0,
                                                     p2, 128, 0, 512, 128, 512, b2, nullptr);
  }

  // 5) recons scores S[b,i,j] = p1[b,i,:]·p2[b,j,:]  (batched M=N=64, K=128)
  {
    int blocks = ((64 / 16) * (64 / 16)) / 4;      // 4
    k_gemm_wmma<<<dim3(blocks, B_), 128, 0, stream>>>(p1, 128, 64 * 128, p2, 128, 64 * 128,
                                                      S, 64, 64 * 64, 64, 64, 128,
                                                      nullptr, nullptr);
  }

  // 6) separable CRF potentials, then jp = crf + log_softmax(S) + prior/n
  k_uv<<<B_ * L_, 128, 0, stream>>>(hbuf, Ws, u, v);
  k_jp<<<B_ * L_, 64, 0, stream>>>(S, u, v, bs, prior, crf, jp);

  // 7) Eisner log-partition and final NLL
  k_eisner<<<B_, 256, 0, stream>>>(crf, logZ);
  k_final<<<1, 32, 0, stream>>>(jp, heads, logZ, (float*)d_out);
}